// ScaledDotProductAttention_53936199303858
// MI455X (gfx1250) — compile-verified
//
#include <hip/hip_runtime.h>

// ---------------------------------------------------------------------------
// Scaled dot-product attention, B=8192, T=8, D=1024, fp32.
// Memory-bound: ~1.03 GB single-touch traffic, ~2 FLOP/byte -> HBM roofline
// floor ~44 us @ 23.3 TB/s. All matrix math via V_WMMA_F32_16X16X4_F32,
// packing 2 batches into one 16x16 tile per wave32. Q/K streamed with
// global_load_b128 + non-temporal hints (pure streaming, no reuse).
//
// K-slot permutation trick: within one 16x16x4 WMMA, which *actual* K index
// occupies each of the 4 hardware K-slots is arbitrary, as long as A and B
// agree. We exploit this so each lane loads 4 CONTIGUOUS floats (b128):
//   lane half h loads Q[row][k0+4h .. k0+4h+3]
//   chunk0 uses elements {0,1} -> slots{0,1}(h=0)/{2,3}(h=1) = K {k0,k0+1,k0+4,k0+5}
//   chunk1 uses elements {2,3} ->                              K {k0+2,k0+3,k0+6,k0+7}
// B (rows of the K/V matrix, lanes = N) uses the identical per-lane fetch, so
// every slot pairs matching K values and the sum covers k0..k0+7 exactly once.
// ---------------------------------------------------------------------------

typedef float v2f __attribute__((ext_vector_type(2)));
typedef float v4f __attribute__((ext_vector_type(4)));
typedef float v8f __attribute__((ext_vector_type(8)));

#define B_DIM 8192
#define T_DIM 8
#define D_DIM 1024
#define NEG_BIG  (-1.0e15f)
#define INV_TEMP 5.0f   // 1 / 0.2

static __device__ __forceinline__ v8f wmma_f32_4(v2f a, v2f b, v8f c) {
    return __builtin_amdgcn_wmma_f32_16x16x4_f32(
        /*neg_a=*/false, a, /*neg_b=*/false, b,
        /*c_mod=*/(short)0, c, /*reuse_a=*/false, /*reuse_b=*/false);
}

__global__ void __launch_bounds__(128)
sdpa_wmma_kernel(const float* __restrict__ q,
                 const float* __restrict__ km,
                 const float* __restrict__ vm,
                 const float* __restrict__ scp,
                 const int*   __restrict__ msk,
                 float* __restrict__ out)
{
    const int tid    = threadIdx.x & 31;       // lane in wave32
    const int wid    = threadIdx.x >> 5;       // wave in block (4 waves/block)
    const int pair   = blockIdx.x * 4 + wid;   // 0..4095, two batches each
    const int rowg   = pair * 2 * T_DIM;       // first global row = b0*8
    const int lane15 = tid & 15;
    const int half   = tid >> 4;               // 0: lanes 0-15, 1: lanes 16-31

    const float scale = scp[0];

    // ---------------- GEMM 1: S = Q * K^T  (K-dim = 1024) -----------------
    const float* qrow = q  + (size_t)(rowg + lane15) * D_DIM + 4 * half;
    const float* krow = km + (size_t)(rowg + lane15) * D_DIM + 4 * half;

    v8f c0 = {0.f,0.f,0.f,0.f,0.f,0.f,0.f,0.f};
    v8f c1 = {0.f,0.f,0.f,0.f,0.f,0.f,0.f,0.f};
    #pragma unroll 4
    for (int k0 = 0; k0 < D_DIM; k0 += 8) {
        v4f a = __builtin_nontemporal_load((const v4f*)(qrow + k0));
        v4f b = __builtin_nontemporal_load((const v4f*)(krow + k0));
        v2f a0 = {a[0], a[1]}, a1 = {a[2], a[3]};
        v2f b0 = {b[0], b[1]}, b1 = {b[2], b[3]};
        c0 = wmma_f32_4(a0, b0, c0);
        c1 = wmma_f32_4(a1, b1, c1);
    }
    v8f s8 = c0 + c1;

    // ---------------- masked softmax (C layout: VGPR r -> rows r / r+8) ----
    // Row m = r + 8*half, col n = lane&15. Valid cells need same batch
    // ((n>>3)==half) and mask[row]&mask[col] nonzero -> lanes 0-7 & 24-31.
    const int  mlane     = msk[rowg + lane15];
    const int  n         = lane15;
    const bool samebatch = ((n >> 3) == half);
    const int  maskn     = __shfl(mlane, n, 32);

    __shared__ float smem[4][256];
    float* sm = smem[wid];
    float* attnOut = out + (size_t)B_DIM * T_DIM * D_DIM;

    #pragma unroll
    for (int r = 0; r < 8; ++r) {
        const int m     = r + 8 * half;
        const int maskm = __shfl(mlane, m, 32);
        float s = s8[r] * scale;
        if (!(samebatch && maskm && maskn)) s = NEG_BIG;   // reference masking
        s *= INV_TEMP;
        float mx = s;
        mx = fmaxf(mx, __shfl_xor(mx, 1, 32));
        mx = fmaxf(mx, __shfl_xor(mx, 2, 32));
        mx = fmaxf(mx, __shfl_xor(mx, 4, 32));
        float p = __expf(s - mx);
        float sum = p;
        sum += __shfl_xor(sum, 1, 32);
        sum += __shfl_xor(sum, 2, 32);
        sum += __shfl_xor(sum, 4, 32);
        p = p / sum;                      // all-masked row -> uniform 1/8 (matches ref)
        if (samebatch) {                  // emit attn output [B,T,T]
            attnOut[(size_t)(rowg + m) * T_DIM + (n & 7)] = p;
        } else {
            p = 0.f;                      // zero cross-batch quadrants for GEMM 2
        }
        sm[m * 16 + n] = p;               // stage for C-layout -> A-layout
    }
    __syncthreads();

    // Reload P in A-layout with the same contiguous-slot permutation:
    // pa4[cp] = P[row][8cp+4h .. 8cp+4h+3]; chunk (cp,j) uses elements {2j,2j+1}
    // -> actual K values {8cp+2j+4h, 8cp+2j+4h+1} in slots {0,1}/{2,3}.
    v4f pa4[2];
    #pragma unroll
    for (int cp = 0; cp < 2; ++cp)
        pa4[cp] = *(const v4f*)(sm + lane15 * 16 + 8 * cp + 4 * half);

    // ---------------- GEMM 2: context = P(16x16) * V(16x1024) -------------
    const float* vbase = vm + (size_t)rowg * D_DIM;
    #pragma unroll 2
    for (int t = 0; t < D_DIM / 16; ++t) {
        const int d0 = t * 16;
        v8f acc = {0.f,0.f,0.f,0.f,0.f,0.f,0.f,0.f};
        #pragma unroll
        for (int cp = 0; cp < 2; ++cp) {
            #pragma unroll
            for (int j = 0; j < 2; ++j) {
                const int kk = 8 * cp + 2 * j + 4 * half;   // B slot rows kk, kk+1
                v2f bch;
                bch[0] = __builtin_nontemporal_load(
                             vbase + (size_t)kk * D_DIM + d0 + lane15);
                bch[1] = __builtin_nontemporal_load(
                             vbase + (size_t)(kk + 1) * D_DIM + d0 + lane15);
                v2f pav = {pa4[cp][2 * j], pa4[cp][2 * j + 1]};
                acc = wmma_f32_4(pav, bch, acc);
            }
        }
        #pragma unroll
        for (int r = 0; r < 8; ++r) {
            const int m = r + 8 * half;
            __builtin_nontemporal_store(
                acc[r], out + (size_t)(rowg + m) * D_DIM + d0 + lane15);
        }
    }
}

extern "C" void kernel_launch(void* const* d_in, const int* in_sizes, int n_in,
                              void* d_out, int out_size, void* d_ws, size_t ws_size,
                              hipStream_t stream) {
    (void)in_sizes; (void)n_in; (void)out_size; (void)d_ws; (void)ws_size;
    const float* q   = (const float*)d_in[0];
    const float* k   = (const float*)d_in[1];
    const float* v   = (const float*)d_in[2];
    const float* sc  = (const float*)d_in[3];
    const int*   m   = (const int*)d_in[4];
    float*       out = (float*)d_out;

    // 4096 batch-pairs, one wave32 each, 4 waves per block.
    dim3 grid(B_DIM / 2 / 4);
    dim3 block(128);
    sdpa_wmma_kernel<<<grid, block, 0, stream>>>(q, k, v, sc, m, out);
}